// SAGEConvN_42717744726727
// MI455X (gfx1250) — compile-verified
//
#include <hip/hip_runtime.h>

#define NFEAT 128
#define NEG_SLOPE 0.2f

typedef __attribute__((ext_vector_type(2))) float v2f;
typedef __attribute__((ext_vector_type(8))) float v8f;

// ---------------------------------------------------------------------------
// Y[nrows x 128] = X[nrows x 128] @ W[128 x 128] (+ Add) (+ Bias broadcast)
// One wave32 computes a full 16x128 output strip: 8 accumulator tiles, so the
// A fragment loaded per k-step feeds 8 v_wmma_f32_16x16x4_f32 ops (8x A reuse,
// no inter-WMMA dependency -> clean back-to-back issue).
// fp32 A 16x4 frag: lane L -> row M = L%16, K pair = (L<16 ? {0,1} : {2,3})
// fp32 B 4x16 frag: lane L -> col N = L%16, same K pair
// fp32 C/D 16x16:   lane L -> col N = L%16, VGPR i -> row M = i + 8*(L>=16)
// ---------------------------------------------------------------------------
__global__ void wmma_gemm_f32(const float* __restrict__ X,
                              const float* __restrict__ W,
                              const float* __restrict__ Add,
                              const float* __restrict__ Bias,
                              float* __restrict__ Y,
                              int nrows)
{
    const int wave = blockIdx.x * (blockDim.x >> 5) + (threadIdx.x >> 5);
    const int lane = threadIdx.x & 31;
    const int nStrips = nrows >> 4;
    if (wave >= nStrips) return;               // wave-uniform guard (EXEC stays full)

    const int m  = lane & 15;                  // row within strip (A) / col within tile (B,C)
    const int kh = (lane >> 4) << 1;           // K offset: 0 or 2

    const float* xrow = X + (size_t)(wave * 16 + m) * NFEAT;

    v8f c[8] = {};                             // 8 column tiles x 8 VGPRs
    for (int k = 0; k < NFEAT; k += 4) {
        v2f a;
        a.x = xrow[k + kh];
        a.y = xrow[k + kh + 1];
        const float* w0 = W + (size_t)(k + kh) * NFEAT + m;
        const float* w1 = w0 + NFEAT;
#pragma unroll
        for (int t = 0; t < 8; ++t) {
            v2f b;
            b.x = w0[t * 16];
            b.y = w1[t * 16];
            c[t] = __builtin_amdgcn_wmma_f32_16x16x4_f32(
                       /*neg_a=*/false, a, /*neg_b=*/false, b,
                       /*c_mod=*/(short)0, c[t], /*reuse_a=*/false, /*reuse_b=*/false);
        }
    }

    const int rowBase = wave * 16 + ((lane >> 4) << 3);
#pragma unroll
    for (int t = 0; t < 8; ++t) {
        const int col = t * 16 + m;
        const float badd = Bias ? Bias[col] : 0.f;
#pragma unroll
        for (int i = 0; i < 8; ++i) {
            size_t off = (size_t)(rowBase + i) * NFEAT + col;
            float v = c[t][i] + badd;
            if (Add) v += Add[off];
            Y[off] = v;
        }
    }
}

// ---------------------------------------------------------------------------
// Per-node attention logits: el_u = h_u.a_l_u, er_u = h_u.a_r_u, same for 'r'.
// One wave per node; 128 feats = one float4 per lane.
// ---------------------------------------------------------------------------
__global__ void node_dots(const float* __restrict__ hu, const float* __restrict__ hr,
                          const float* __restrict__ alu, const float* __restrict__ aru,
                          const float* __restrict__ alr, const float* __restrict__ arr,
                          float* __restrict__ elu, float* __restrict__ eru,
                          float* __restrict__ elr, float* __restrict__ err_, int n)
{
    const int wave = blockIdx.x * (blockDim.x >> 5) + (threadIdx.x >> 5);
    const int lane = threadIdx.x & 31;
    if (wave >= n) return;
    const float4 xu  = ((const float4*)(hu + (size_t)wave * NFEAT))[lane];
    const float4 xr  = ((const float4*)(hr + (size_t)wave * NFEAT))[lane];
    const float4 vlu = ((const float4*)alu)[lane];
    const float4 vru = ((const float4*)aru)[lane];
    const float4 vlr = ((const float4*)alr)[lane];
    const float4 vrr = ((const float4*)arr)[lane];
    float s0 = xu.x*vlu.x + xu.y*vlu.y + xu.z*vlu.z + xu.w*vlu.w;
    float s1 = xu.x*vru.x + xu.y*vru.y + xu.z*vru.z + xu.w*vru.w;
    float s2 = xr.x*vlr.x + xr.y*vlr.y + xr.z*vlr.z + xr.w*vlr.w;
    float s3 = xr.x*vrr.x + xr.y*vrr.y + xr.z*vrr.z + xr.w*vrr.w;
#pragma unroll
    for (int off = 16; off > 0; off >>= 1) {
        s0 += __shfl_down(s0, off);
        s1 += __shfl_down(s1, off);
        s2 += __shfl_down(s2, off);
        s3 += __shfl_down(s3, off);
    }
    if (lane == 0) { elu[wave] = s0; eru[wave] = s1; elr[wave] = s2; err_[wave] = s3; }
}

__global__ void fill_f32(float* __restrict__ p, float v, size_t n)
{
    size_t i = (size_t)blockIdx.x * blockDim.x + threadIdx.x;
    const size_t stride = (size_t)gridDim.x * blockDim.x;
    for (; i < n; i += stride) p[i] = v;
}

__device__ inline void atomicMaxF32(float* addr, float val)
{
    unsigned int* ai = (unsigned int*)addr;
    unsigned int cur = *ai;
    while (__uint_as_float(cur) < val) {
        unsigned int prev = atomicCAS(ai, cur, __float_as_uint(val));
        if (prev == cur) break;
        cur = prev;
    }
}

// Pass 1: per-dst running max of leaky_relu(el[src]+er[dst])
__global__ void edge_max(const int* __restrict__ src, const int* __restrict__ dst,
                         const float* __restrict__ el, const float* __restrict__ er,
                         float* __restrict__ m, int ne)
{
    int i = blockIdx.x * blockDim.x + threadIdx.x;
    if (i >= ne) return;
    float e = el[src[i]] + er[dst[i]];
    e = (e > 0.f) ? e : NEG_SLOPE * e;
    atomicMaxF32(&m[dst[i]], e);
}

// Pass 2: scatter exp(e-m) and exp(e-m)*h[src] into per-dst accumulators.
// One wave per edge; 128 feats = one float4 per lane.
__global__ void edge_scatter(const int* __restrict__ src, const int* __restrict__ dst,
                             const float* __restrict__ el, const float* __restrict__ er,
                             const float* __restrict__ m,  const float* __restrict__ h,
                             float* __restrict__ neigh, float* __restrict__ denom, int ne)
{
    const int wave = blockIdx.x * (blockDim.x >> 5) + (threadIdx.x >> 5);
    const int lane = threadIdx.x & 31;
    if (wave >= ne) return;
    const int s = src[wave], d = dst[wave];
    float e = el[s] + er[d];
    e = (e > 0.f) ? e : NEG_SLOPE * e;
    const float ex = __expf(e - m[d]);
    if (lane == 0) atomicAdd(&denom[d], ex);
    const float4 hv = ((const float4*)(h + (size_t)s * NFEAT))[lane];
    float* nd = neigh + (size_t)d * NFEAT + lane * 4;
    atomicAdd(nd + 0, hv.x * ex);
    atomicAdd(nd + 1, hv.y * ex);
    atomicAdd(nd + 2, hv.z * ex);
    atomicAdd(nd + 3, hv.w * ex);
}

// neigh = 0.5 * (acc_u/denom_u + acc_r/denom_r), 0 for isolated dst nodes
__global__ void normalize_neigh(const float* __restrict__ nu, const float* __restrict__ nr,
                                const float* __restrict__ du, const float* __restrict__ dr,
                                float* __restrict__ out, size_t total)
{
    size_t idx = (size_t)blockIdx.x * blockDim.x + threadIdx.x;
    if (idx >= total) return;
    const size_t node = idx >> 7;                  // F = 128
    const float duv = du[node], drv = dr[node];
    const float a = (duv > 0.f) ? nu[idx] / duv : 0.f;
    const float b = (drv > 0.f) ? nr[idx] / drv : 0.f;
    out[idx] = 0.5f * (a + b);
}

extern "C" void kernel_launch(void* const* d_in, const int* in_sizes, int n_in,
                              void* d_out, int out_size, void* d_ws, size_t ws_size,
                              hipStream_t stream)
{
    const float* feat    = (const float*)d_in[0];
    const int*   src_u   = (const int*)d_in[1];
    const int*   dst_u   = (const int*)d_in[2];
    const int*   src_r   = (const int*)d_in[3];
    const int*   dst_r   = (const int*)d_in[4];
    const float* W_self  = (const float*)d_in[5];
    const float* W_u     = (const float*)d_in[6];
    const float* a_l_u   = (const float*)d_in[7];
    const float* a_r_u   = (const float*)d_in[8];
    const float* W_r     = (const float*)d_in[9];
    const float* a_l_r   = (const float*)d_in[10];
    const float* a_r_r   = (const float*)d_in[11];
    const float* W_neigh = (const float*)d_in[12];
    const float* b_neigh = (const float*)d_in[13];
    const float* bias    = (const float*)d_in[14];
    float* out = (float*)d_out;

    const int N = in_sizes[0] / NFEAT;
    const int E = in_sizes[1];
    const size_t NF = (size_t)N * NFEAT;

    float* ws      = (float*)d_ws;
    float* h_u     = ws;             // 51.2 MB; later reused for combined neigh
    float* h_r     = ws + NF;        // 51.2 MB; later reused for h_neigh
    float* neigh_u = ws + 2 * NF;
    float* neigh_r = ws + 3 * NF;
    float* elu     = ws + 4 * NF;
    float* eru     = elu + N;
    float* elr     = elu + 2 * (size_t)N;
    float* err_    = elu + 3 * (size_t)N;
    float* m_u     = elu + 4 * (size_t)N;
    float* m_r     = elu + 5 * (size_t)N;
    float* den_u   = elu + 6 * (size_t)N;
    float* den_r   = elu + 7 * (size_t)N;

    const int strips = N / 16;                      // 16-row output strips
    const int gemmBl = (strips + 3) / 4;            // 4 waves (128 thr) per block

    // Projections h = feat @ W (WMMA fp32)
    wmma_gemm_f32<<<gemmBl, 128, 0, stream>>>(feat, W_u, nullptr, nullptr, h_u, N);
    wmma_gemm_f32<<<gemmBl, 128, 0, stream>>>(feat, W_r, nullptr, nullptr, h_r, N);

    // Attention logits
    node_dots<<<(N + 7) / 8, 256, 0, stream>>>(h_u, h_r, a_l_u, a_r_u, a_l_r, a_r_r,
                                               elu, eru, elr, err_, N);

    // Init accumulators
    fill_f32<<<2048, 256, 0, stream>>>(neigh_u, 0.f, 2 * NF);          // neigh_u + neigh_r
    fill_f32<<<256, 256, 0, stream>>>(den_u, 0.f, (size_t)2 * N);      // den_u + den_r
    fill_f32<<<256, 256, 0, stream>>>(m_u, -1e30f, (size_t)2 * N);     // m_u + m_r

    // Edge softmax pass 1: segment max
    edge_max<<<(E + 255) / 256, 256, 0, stream>>>(src_u, dst_u, elu, eru, m_u, E);
    edge_max<<<(E + 255) / 256, 256, 0, stream>>>(src_r, dst_r, elr, err_, m_r, E);

    // Edge softmax pass 2: unnormalized scatter of exp(e-m) and exp(e-m)*h[src]
    edge_scatter<<<(E + 7) / 8, 256, 0, stream>>>(src_u, dst_u, elu, eru, m_u,
                                                  h_u, neigh_u, den_u, E);
    edge_scatter<<<(E + 7) / 8, 256, 0, stream>>>(src_r, dst_r, elr, err_, m_r,
                                                  h_r, neigh_r, den_r, E);

    // neigh = 0.5*(nu/du + nr/dr)  -> reuse h_u buffer
    normalize_neigh<<<(int)((NF + 255) / 256), 256, 0, stream>>>(neigh_u, neigh_r,
                                                                 den_u, den_r, h_u, NF);

    // h_neigh = neigh @ W_neigh + b_neigh  -> reuse h_r buffer
    wmma_gemm_f32<<<gemmBl, 128, 0, stream>>>(h_u, W_neigh, nullptr, b_neigh, h_r, N);

    // out = feat @ W_self + h_neigh + bias
    wmma_gemm_f32<<<gemmBl, 128, 0, stream>>>(feat, W_self, h_r, bias, out, N);
}